// fCGModule_16243566313950
// MI455X (gfx1250) — compile-verified
//
#include <hip/hip_runtime.h>
#include <hip/hip_bf16.h>

// ---------------- compile-time CG-product tables ----------------
constexpr int MAXL = 5;
constexpr int TAU  = 12;

constexpr int imin_(int a, int b) { return a < b ? a : b; }

constexpr int countTrip() {
  int c = 0;
  for (int l1 = 0; l1 <= MAXL; l1++)
    for (int l2 = 0; l2 <= l1; l2++)
      for (int l = l1 - l2; l <= imin_(l1 + l2, MAXL); l++) c++;
  return c;
}
constexpr int NTRIP = countTrip();

constexpr int countCoef() {
  int c = 0;
  for (int l1 = 0; l1 <= MAXL; l1++)
    for (int l2 = 0; l2 <= l1; l2++)
      for (int l = l1 - l2; l <= imin_(l1 + l2, MAXL); l++)
        c += (2 * l + 1) * (2 * l1 + 1);
  return c;
}
constexpr int NCOEF = countCoef();

constexpr int countJobs() {
  int c = 0;
  for (int l1 = 0; l1 <= MAXL; l1++)
    for (int l2 = 0; l2 <= l1; l2++)
      for (int l = l1 - l2; l <= imin_(l1 + l2, MAXL); l++)
        c += (2 * TAU * (2 * l + 1) + 15) / 16;   // N tiles of 16 (N = 24*(2l+1))
  return c;
}
constexpr int NJOBS = countJobs();

struct Tables {
  int l1[NTRIP], l2[NTRIP], ll[NTRIP];
  int coefOff[NTRIP];            // float offset into coef table
  int inA[NTRIP], inB[NTRIP];    // complex offset of l1/l2 block within batch
  long long outF[NTRIP];         // float offset of fragment within batch output
  short jobT[NJOBS];
  short jobTile[NJOBS];
  long long totalOutF;
};

constexpr Tables mkTables() {
  Tables T{};
  int cum[MAXL + 2] = {};
  for (int l = 0; l <= MAXL; l++) cum[l + 1] = cum[l] + TAU * (2 * l + 1);
  int cnt[MAXL + 1] = {};
  int gi[NTRIP] = {};
  int t = 0, co = 0;
  for (int l1 = 0; l1 <= MAXL; l1++)
    for (int l2 = 0; l2 <= l1; l2++)
      for (int l = l1 - l2; l <= imin_(l1 + l2, MAXL); l++) {
        T.l1[t] = l1; T.l2[t] = l2; T.ll[t] = l;
        T.coefOff[t] = co; co += (2 * l + 1) * (2 * l1 + 1);
        T.inA[t] = cum[l1]; T.inB[t] = cum[l2];
        gi[t] = cnt[l]++;
        t++;
      }
  long long base[MAXL + 1] = {};
  long long acc = 0;
  for (int l = 0; l <= MAXL; l++) { base[l] = acc; acc += (long long)2 * TAU * TAU * (2 * l + 1) * cnt[l]; }
  T.totalOutF = acc;
  for (int q = 0; q < NTRIP; q++)
    T.outF[q] = base[T.ll[q]] + (long long)gi[q] * 2 * TAU * TAU * (2 * T.ll[q] + 1);
  int j = 0;
  for (int q = 0; q < NTRIP; q++) {
    int nt = (2 * TAU * (2 * T.ll[q] + 1) + 15) / 16;
    for (int tt = 0; tt < nt; tt++) { T.jobT[j] = (short)q; T.jobTile[j] = (short)tt; j++; }
  }
  return T;
}

constexpr Tables HT = mkTables();
static_assert(HT.totalOutF == 131616, "per-batch output float count mismatch");
__constant__ Tables gT = mkTables();

constexpr int TOTAL_IN = TAU * 36;   // 432 complex per batch
constexpr long long TOTAL_OUT_F = 131616;

// ---------------- CG coefficient setup kernel (runs every launch) ----------------
__device__ double dfact(int n) {
  double r = 1.0;
  for (int i = 2; i <= n; i++) r *= (double)i;
  return r;
}

__global__ void cg_setup(float* __restrict__ coefs) {
  int idx = blockIdx.x * blockDim.x + threadIdx.x;
  if (idx >= NCOEF) return;
  int t = 0;
  while (t + 1 < NTRIP && gT.coefOff[t + 1] <= idx) t++;
  int l1 = gT.l1[t], l2 = gT.l2[t], l = gT.ll[t];
  int local = idx - gT.coefOff[t];
  int w1 = 2 * l1 + 1;
  int k  = local / w1;
  int mi = local % w1;
  int m1 = mi - l1;
  int m  = k - l;
  int m2 = m - m1;
  float out = 0.0f;
  if (m2 >= -l2 && m2 <= l2) {
    double pref = sqrt((double)(2 * l + 1) * dfact(l + l1 - l2) * dfact(l - l1 + l2) *
                       dfact(l1 + l2 - l) / dfact(l1 + l2 + l + 1));
    pref *= sqrt(dfact(l + m) * dfact(l - m) * dfact(l1 - m1) * dfact(l1 + m1) *
                 dfact(l2 - m2) * dfact(l2 + m2));
    int kmin = max(0, max(l2 - l - m1, l1 + m2 - l));
    int kmax = min(l1 + l2 - l, min(l1 - m1, l2 + m2));
    double s = 0.0;
    for (int kk = kmin; kk <= kmax; kk++) {
      double d = dfact(kk) * dfact(l1 + l2 - l - kk) * dfact(l1 - m1 - kk) *
                 dfact(l2 + m2 - kk) * dfact(l - l2 + m1 + kk) * dfact(l - l1 - m2 + kk);
      s += ((kk & 1) ? -1.0 : 1.0) / d;
    }
    out = (float)(pref * s);
  }
  coefs[idx] = out;
}

// ---------------- main WMMA kernel ----------------
typedef __attribute__((ext_vector_type(2))) float v2f;
typedef __attribute__((ext_vector_type(8))) float v8f;

constexpr int WAVES_PER_BLOCK = 8;
constexpr int YSPLIT = 4;

__launch_bounds__(256)
__global__ void cg_main(const float* __restrict__ act,
                        float* __restrict__ out,
                        const float* __restrict__ coefs) {
  __shared__ float2 sAct[TOTAL_IN];
  __shared__ float  sCoef[NCOEF];

  const int b   = blockIdx.x;
  const int tid = threadIdx.x;

  const float2* actB = (const float2*)act + (size_t)b * TOTAL_IN;
  for (int i = tid; i < TOTAL_IN; i += 256) sAct[i] = actB[i];
  for (int i = tid; i < NCOEF;    i += 256) sCoef[i] = coefs[i];
  __syncthreads();

  const int wave = tid >> 5;
  const int lane = tid & 31;
  const int nloc = lane & 15;   // N column within half-wave; also A row M
  const int grp  = lane >> 4;   // half-wave select
  const bool aRowOK = nloc < TAU;
  const int aRowIdx = aRowOK ? nloc : 0;

  float* outB = out + (size_t)b * TOTAL_OUT_F;

  for (int job0 = blockIdx.y * WAVES_PER_BLOCK + wave; job0 < NJOBS;
       job0 += WAVES_PER_BLOCK * YSPLIT) {
    // job0 is wave-uniform by construction; make it an SGPR so all triple
    // metadata becomes scalar loads / scalar control flow.
    const int job  = __builtin_amdgcn_readfirstlane(job0);
    const int t    = gT.jobT[job];
    const int tile = gT.jobTile[job];
    const int l1 = gT.l1[t], l2 = gT.l2[t], l = gT.ll[t];
    const int w1 = 2 * l1 + 1, w2 = 2 * l2 + 1, wl = 2 * l + 1;
    const int Ntot = 2 * TAU * wl;            // floats per output row (i)
    const int col  = tile * 16 + nloc;        // output column (j,k,c flattened)
    const bool cvalid = col < Ntot;

    const int ccol = col & 1;
    const int jk   = col >> 1;
    int kk = jk % wl;
    int jj = jk / wl;
    if (!cvalid) { kk = 0; jj = 0; }
    const int nbase = kk - l + l1 + l2;       // nidx = nbase - m (unique n with C!=0)

    const float2* aRow  = sAct + gT.inA[t] + aRowIdx * w1;
    const float2* bRow  = sAct + gT.inB[t] + jj * w2;
    const float*  cfRow = sCoef + gT.coefOff[t] + kk * w1;

    v8f acc = {0.f, 0.f, 0.f, 0.f, 0.f, 0.f, 0.f, 0.f};

#pragma unroll
    for (int ks = 0; ks < MAXL + 1; ks++) {
      if (ks > l1) break;                     // scalar (l1 in SGPR) -> uniform branch
      const int  m   = 2 * ks + grp;          // K pair index; cc==v for the 2 lanes' regs
      const bool mok = m < w1;
      const int  mC  = mok ? m : 0;

      // A fragment: one complex value covers both regs (v=0 -> re, v=1 -> im)
      const float2 a = aRow[mC];
      const float aMask = (aRowOK & mok) ? 1.0f : 0.0f;
      v2f A;
      A[0] = a.x * aMask;
      A[1] = a.y * aMask;

      // B fragment: T[(j,k),m] = coef * b[j, nbase-m]
      const int  nidx = nbase - m;
      const bool bok  = cvalid & mok & (nidx >= 0) & (nidx < w2);
      const float2 bv = bRow[bok ? nidx : 0];  // clamped address, unconditional load
      const float cfv = cfRow[mC];             // unconditional load
      const float g   = bok ? cfv : 0.0f;
      // complex product as real GEMM: re col: [T_re; -T_im], im col: [T_im; T_re]
      const float b0 = ccol ? bv.y : bv.x;     // row cc=0 (pairs with a_re)
      const float b1 = ccol ? bv.x : -bv.y;    // row cc=1 (pairs with a_im)
      v2f B;
      B[0] = g * b0;
      B[1] = g * b1;

      acc = __builtin_amdgcn_wmma_f32_16x16x4_f32(
          /*neg_a=*/false, A, /*neg_b=*/false, B,
          /*c_mod=*/(short)0, acc, /*reuse_a=*/false, /*reuse_b=*/false);
    }

    if (cvalid) {
      float* frag = outB + gT.outF[t] + col;
      // rows r + 8*grp for r<4 are always < 12 -> unconditional
#pragma unroll
      for (int r = 0; r < 4; r++)
        frag[(long long)(r + 8 * grp) * Ntot] = acc[r];
      if (grp == 0) {
#pragma unroll
        for (int r = 4; r < 8; r++)
          frag[(long long)r * Ntot] = acc[r];
      }
    }
  }
}

// ---------------- launch ----------------
extern "C" void kernel_launch(void* const* d_in, const int* in_sizes, int n_in,
                              void* d_out, int out_size, void* d_ws, size_t ws_size,
                              hipStream_t stream) {
  const float* act = (const float*)d_in[0];
  float* out   = (float*)d_out;
  float* coefs = (float*)d_ws;

  const int B = in_sizes[0] / (TOTAL_IN * 2);

  cg_setup<<<(NCOEF + 255) / 256, 256, 0, stream>>>(coefs);

  dim3 grid(B, YSPLIT);
  cg_main<<<grid, 256, 0, stream>>>(act, out, coefs);
}